// GraphConstruction_53017076302013
// MI455X (gfx1250) — compile-verified
//
#include <hip/hip_runtime.h>
#include <hip/hip_bf16.h>

typedef float v2f __attribute__((ext_vector_type(2)));
typedef float v8f __attribute__((ext_vector_type(8)));

#define NPATCH 4096
#define NTILE  256                   // NPATCH / 16
#define IMG    1024
#define CHUNKS 4                     // K=16 as 4 chunks of K=4 (sq chunk synthesized)
#define FRAG_FLOATS (CHUNKS * 64)    // 256 floats per (tile, col)
#define TILE_FLOATS (16 * FRAG_FLOATS)   // 4096 floats per output tile
#define TILE_BYTES  (TILE_FLOATS * 4)    // 16 KB staged per J-tile
#define BUF_FLOATS  (TILE_FLOATS + 512)  // + padded SQ^T region (bank/OOB safe)

// patches[n, a, b] under the reference's row-major reshape of the unfold output:
//   n = ki*256 + kj*16 + (bi>>2);  a = (bi&3)*4 + (bj>>4);  b = bj & 15
__device__ __forceinline__ float patch_elem(const float* __restrict__ x, int n, int a, int b) {
    int ki = n >> 8;
    int kj = (n >> 4) & 15;
    int bh = n & 15;
    int bi = bh * 4 + (a >> 2);
    int bj = (a & 3) * 16 + b;
    int y  = bi * 16 + ki;
    int xc = bj * 16 + kj;
    return x[y * IMG + xc];
}

// SQ[n*16 + b] = sum_a patches[n,a,b]^2
__global__ void GraphConstruction_sq_kernel(const float* __restrict__ x, float* __restrict__ sq) {
    int tid = blockIdx.x * blockDim.x + threadIdx.x;   // [0, 65536)
    int n = tid >> 4, b = tid & 15;
    float s = 0.f;
    #pragma unroll
    for (int a = 0; a < 16; ++a) { float p = patch_elem(x, n, a, b); s += p * p; }
    sq[tid] = s;
}

// WMMA fragment images, 4 K-chunks (A pre-scaled by -2):
//   idx = (((t*16 + b)*4 + c)*32 + lane)*2 + v
//   m/n = t*16 + (lane&15);  klocal = 2*(lane>=16) + v;  K = 4c + klocal
__global__ void GraphConstruction_frag_kernel(const float* __restrict__ x,
                                              float* __restrict__ PA,
                                              float* __restrict__ PB) {
    int idx = blockIdx.x * blockDim.x + threadIdx.x;   // [0, 256*16*4*64)
    int v  = idx & 1;
    int l  = (idx >> 1) & 31;
    int c  = (idx >> 6) & 3;
    int tb = idx >> 8;             // t*16 + b
    int b  = tb & 15;
    int t  = tb >> 4;
    int m  = t * 16 + (l & 15);
    int klocal = ((l >> 4) << 1) + v;
    float p = patch_elem(x, m, 4 * c + klocal, b);
    PA[idx] = -2.0f * p;
    PB[idx] = p;
}

// Async DMA one J-tile (16 KB B frags + 1 KB transposed SQ row) into LDS buffer.
__device__ __forceinline__ void stage_tile(int t, int J, uint32_t lds_base,
                                           uint64_t pb_base, uint64_t sq_base) {
    #pragma unroll
    for (int i = 0; i < 4; ++i) {
        uint32_t f4 = (uint32_t)t + i * 256u;
        uint32_t g  = (uint32_t)J * TILE_BYTES + f4 * 16u;
        uint32_t l  = lds_base + f4 * 16u;
        asm volatile("global_load_async_to_lds_b128 %0, %1, %2"
                     :: "v"(l), "v"(g), "s"(pb_base) : "memory");
    }
    uint32_t g = (((uint32_t)J * 16u + (t & 15)) * 16u + (t >> 4)) * 4u;
    uint32_t l = lds_base + (uint32_t)TILE_BYTES + (((t >> 4) * 16u) + (t & 15)) * 4u;
    asm volatile("global_load_async_to_lds_b32 %0, %1, %2"
                 :: "v"(l), "v"(g), "s"(sq_base) : "memory");
}

// 8 waves/WG; wave w owns I-tile = blockIdx.y*8 + w with A fragments resident
// in VGPRs; sweeps 32 J-tiles with double-buffered async LDS staging.
__global__ __launch_bounds__(256) void GraphConstruction_gram_kernel(
        const float* __restrict__ PA, const float* __restrict__ PB,
        const float* __restrict__ SQ, float* __restrict__ out) {
    __shared__ float Smem[2 * BUF_FLOATS];             // 2 x 18 KB
    const int t    = threadIdx.x;
    const int lane = t & 31;
    const int wave = t >> 5;
    const int I  = blockIdx.y * 8 + wave;              // gridDim.y = 32
    const int j0 = blockIdx.x * 32;                    // gridDim.x = 8

    const uint32_t lds0 = (uint32_t)(uintptr_t)(&Smem[0]);
    const uint32_t lds1 = (uint32_t)(uintptr_t)(&Smem[BUF_FLOATS]);

    // Masked one-vector + per-column A-side sq: synthesize the 5th K-chunk
    // A' = [sq_i, 1, 0, 0] / B' = [1, sq_j, 0, 0] so acc == d2 directly.
    const float onev = (lane < 16) ? 1.0f : 0.0f;
    float sqa[16];
    #pragma unroll
    for (int b = 0; b < 16; ++b)
        sqa[b] = (lane < 16) ? SQ[(size_t)(I * 16 + lane) * 16 + b] : 0.0f;

    // Resident A fragments: 16 cols x 4 chunks x v2f = 128 VGPRs.
    v2f afr[16][CHUNKS];
    #pragma unroll
    for (int b = 0; b < 16; ++b) {
        #pragma unroll
        for (int c = 0; c < CHUNKS; ++c)
            afr[b][c] = *(const v2f*)(PA + (((size_t)(I * 16 + b) * CHUNKS + c) * 64) + 2 * lane);
    }

    const uint64_t pb_base = (uint64_t)(uintptr_t)PB;
    const uint64_t sq_base = (uint64_t)(uintptr_t)SQ;

    // Prologue: stage first tile.
    stage_tile(t, j0, lds0, pb_base, sq_base);
    asm volatile("s_wait_asynccnt 0x0" ::: "memory");
    __syncthreads();

    int buf = 0;
    #pragma unroll 1
    for (int jj = 0; jj < 32; ++jj) {
        const int J = j0 + jj;
        const float* Bsc = &Smem[buf ? BUF_FLOATS : 0];

        // Prefetch next tile into the other buffer while we compute this one.
        if (jj < 31)
            stage_tile(t, J + 1, buf ? lds0 : lds1, pb_base, sq_base);

        v8f dmax = {0.f, 0.f, 0.f, 0.f, 0.f, 0.f, 0.f, 0.f};
        #pragma unroll
        for (int b = 0; b < 16; ++b) {
            v2f bf[CHUNKS];
            #pragma unroll
            for (int c = 0; c < CHUNKS; ++c)
                bf[c] = *(const v2f*)(Bsc + (b * CHUNKS + c) * 64 + 2 * lane);
            const float sqj = Bsc[TILE_FLOATS + b * 16 + lane] * onev;  // mask, no divergence

            v8f acc = {0.f, 0.f, 0.f, 0.f, 0.f, 0.f, 0.f, 0.f};
            #pragma unroll
            for (int c = 0; c < CHUNKS; ++c)
                acc = __builtin_amdgcn_wmma_f32_16x16x4_f32(
                        false, afr[b][c], false, bf[c], (short)0, acc, false, false);
            v2f a4 = {sqa[b], onev};
            v2f b4 = {onev, sqj};
            acc = __builtin_amdgcn_wmma_f32_16x16x4_f32(
                    false, a4, false, b4, (short)0, acc, false, false);

            #pragma unroll
            for (int r = 0; r < 8; ++r) dmax[r] = fmaxf(dmax[r], acc[r]);

            // Bound scheduler lookahead: keep <=2 column accumulators live.
            if ((b & 1) == 1) __builtin_amdgcn_sched_barrier(0x100);  // DS-reads may cross
        }

        // C/D layout: vgpr r -> row M = r + 8*(lane>=16), col N = lane&15
        const int colBase = J * 16 + (lane & 15);
        const int rowBase = I * 16 + ((lane >> 4) << 3);
        #pragma unroll
        for (int r = 0; r < 8; ++r)
            __builtin_nontemporal_store((dmax[r] <= 49.0f) ? 1.0f : 0.0f,
                                        &out[(size_t)(rowBase + r) * NPATCH + colBase]);

        asm volatile("s_wait_asynccnt 0x0" ::: "memory");
        __syncthreads();
        buf ^= 1;
    }
}

extern "C" void kernel_launch(void* const* d_in, const int* in_sizes, int n_in,
                              void* d_out, int out_size, void* d_ws, size_t ws_size,
                              hipStream_t stream) {
    const float* x = (const float*)d_in[0];
    float* out = (float*)d_out;

    float* PA = (float*)d_ws;                                   // 4 MB
    float* PB = PA + (size_t)NTILE * TILE_FLOATS;               // 4 MB
    float* SQ = PB + (size_t)NTILE * TILE_FLOATS;               // 256 KB

    GraphConstruction_sq_kernel<<<256, 256, 0, stream>>>(x, SQ);
    GraphConstruction_frag_kernel<<<(NTILE * TILE_FLOATS) / 256, 256, 0, stream>>>(x, PA, PB);

    dim3 grid(8, 32);
    GraphConstruction_gram_kernel<<<grid, 256, 0, stream>>>(PA, PB, SQ, out);
}